// _IPEXAttention_7000796692985
// MI455X (gfx1250) — compile-verified
//
#include <hip/hip_runtime.h>
#include <math.h>

#define B_     16
#define KVLEN_ 4096
#define BLOCK_ 64
#define H_     32
#define KVH_   8
#define D_     128
#define HID_   4096
#define G_     4
#define NTILES 256            // KVLEN_/16
#define SCALE_ 0.08838834764831845f   // 1/sqrt(128)

typedef float v2f __attribute__((ext_vector_type(2)));
typedef float v8f __attribute__((ext_vector_type(8)));

__device__ __forceinline__ v8f wmma_f32_4(v2f a, v2f b, v8f c) {
  // D = A(16x4 f32) * B(4x16 f32) + C(16x16 f32)
  return __builtin_amdgcn_wmma_f32_16x16x4_f32(false, a, false, b, (short)0, c,
                                               false, false);
}

// ---------------------------------------------------------------------------
// Kernel 1: fused QKV projection.  y = x @ [wq|wk|wv], M=16 rows (= batch).
// One wave per 16-wide N tile.  6144/16 = 384 tiles.
// ---------------------------------------------------------------------------
__global__ __launch_bounds__(32)
void qkv_kernel(const float* __restrict__ x,
                const float* __restrict__ wq,
                const float* __restrict__ wk,
                const float* __restrict__ wv,
                float* __restrict__ qo,
                float* __restrict__ ko,
                float* __restrict__ vo) {
  const int lane = threadIdx.x & 31;
  const int m16  = lane & 15;        // M row (batch)
  const int half = lane >> 4;
  const int koff = half << 1;        // K offset inside 4-chunk

  const int n0 = blockIdx.x * 16;
  const float* W; int ldw; int col; float* outp; int ldo;
  if (n0 < H_ * D_)                { W = wq; ldw = H_ * D_;  col = n0 + m16;                     outp = qo; ldo = H_ * D_; }
  else if (n0 < (H_ + KVH_) * D_)  { W = wk; ldw = KVH_ * D_; col = n0 - H_ * D_ + m16;          outp = ko; ldo = KVH_ * D_; }
  else                             { W = wv; ldw = KVH_ * D_; col = n0 - (H_ + KVH_) * D_ + m16; outp = vo; ldo = KVH_ * D_; }

  const float* xr = x + m16 * HID_;
  v8f acc = {0.f, 0.f, 0.f, 0.f, 0.f, 0.f, 0.f, 0.f};

  for (int kb = 0; kb < HID_; kb += 32) {
    if (kb + 128 < HID_)
      __builtin_prefetch(&W[(size_t)(kb + 128) * ldw + col], 0, 1);
#pragma unroll
    for (int kc = 0; kc < 32; kc += 4) {
      const int k = kb + kc + koff;
      v2f a = *(const v2f*)(xr + k);
      v2f b;
      b.x = W[(size_t)k * ldw + col];
      b.y = W[(size_t)(k + 1) * ldw + col];
      acc = wmma_f32_4(a, b, acc);
    }
  }

#pragma unroll
  for (int g = 0; g < 8; ++g) {
    const int row = g + 8 * half;                 // batch row
    outp[row * ldo + col] = acc[g];
  }
}

// ---------------------------------------------------------------------------
// Kernel 2: rotate-half RoPE in place on q (B,H,D) and new-k (B,KVH,D).
// ---------------------------------------------------------------------------
__global__ __launch_bounds__(64)
void rope_kernel(float* __restrict__ q, float* __restrict__ kn,
                 const int* __restrict__ input_lens) {
  const int idx = blockIdx.x;
  const int b = idx / (H_ + KVH_);
  const int h = idx % (H_ + KVH_);
  const int j = threadIdx.x;        // 0..63
  float* p = (h < H_) ? (q + (b * H_ + h) * D_)
                      : (kn + (b * KVH_ + (h - H_)) * D_);
  const float pos = (float)input_lens[b];
  const float inv_freq = __powf(10000.0f, -((float)j) / 64.0f);
  const float ang = pos * inv_freq;
  float s, c;
  __sincosf(ang, &s, &c);
  const float x1 = p[j];
  const float x2 = p[j + 64];
  p[j]      = x1 * c - x2 * s;
  p[j + 64] = x2 * c + x1 * s;
}

// ---------------------------------------------------------------------------
// Kernel 3: GQA flash-decode.  One workgroup per (b, kv-head), 8 waves stride
// over 256 key tiles of 16.  q group (4 rows) padded to M=16.  WMMA for both
// q@K^T and P@V; online softmax per wave; LDS combine across waves.
// ---------------------------------------------------------------------------
__device__ __forceinline__ const float* kv_row(const float* __restrict__ cache,
                                               const float* __restrict__ newrow,
                                               const int* __restrict__ bt,
                                               int t, int pos, int h) {
  if (t == pos) return newrow;
  const int blk = bt[t >> 6];
  const int off = t & 63;
  return cache + (size_t)((blk * KVH_ + h) * BLOCK_ + off) * D_;
}

__global__ __launch_bounds__(256)
void attn_kernel(const float* __restrict__ q,
                 const float* __restrict__ kn,
                 const float* __restrict__ vn,
                 const float* __restrict__ kcache,
                 const float* __restrict__ vcache,
                 const int* __restrict__ block_tables,
                 const int* __restrict__ input_lens,
                 float* __restrict__ out) {
  __shared__ float ldsP[8][16][16];   // per-wave P relayout scratch
  __shared__ float ldsO[8][4][128];   // per-wave partial outputs
  __shared__ float ldsM[8][4];
  __shared__ float ldsL[8][4];

  const int bh   = blockIdx.x;
  const int b    = bh / KVH_;
  const int h    = bh % KVH_;
  const int tid  = threadIdx.x;
  const int w    = tid >> 5;
  const int lane = tid & 31;
  const int m16  = lane & 15;
  const int half = lane >> 4;
  const int koff = half << 1;

  const int  pos = input_lens[b];
  const int* bt  = block_tables + b * (KVLEN_ / BLOCK_);
  const float* knrow = kn + (b * KVH_ + h) * D_;
  const float* vnrow = vn + (b * KVH_ + h) * D_;

  // q A-fragments for all 32 K-chunks (rows >= G_ padded with zeros)
  v2f qa[32];
  {
    const float* qrow = q + (b * H_ + h * G_ + m16) * D_;
#pragma unroll
    for (int c = 0; c < 32; ++c) {
      if (m16 < G_) qa[c] = *(const v2f*)(qrow + 4 * c + koff);
      else          { qa[c].x = 0.f; qa[c].y = 0.f; }
    }
  }

  const v8f vzero = {0.f, 0.f, 0.f, 0.f, 0.f, 0.f, 0.f, 0.f};
  float mrow[4] = {-INFINITY, -INFINITY, -INFINITY, -INFINITY};
  float lrow[4] = {0.f, 0.f, 0.f, 0.f};
  v8f acc[8];
#pragma unroll
  for (int nt = 0; nt < 8; ++nt) acc[nt] = vzero;

  for (int tile = w; tile < NTILES; tile += 8) {
    const int t0 = tile * 16;
    if (t0 > pos) continue;                       // fully masked tile

    // ---- S = q @ K^T (16x16 tile), K-loop over D in chunks of 4 ----
    const int tcol = t0 + m16;                    // this lane's key column
    const float* krow = kv_row(kcache, knrow, bt, tcol, pos, h);
    v8f S = vzero;
#pragma unroll
    for (int c = 0; c < 32; ++c) {
      v2f kb = *(const v2f*)(krow + 4 * c + koff);
      S = wmma_f32_4(qa[c], kb, S);
    }

    // ---- online softmax on the 4 real rows ----
    const bool valid = (tcol <= pos);
    float pvals[8] = {0.f, 0.f, 0.f, 0.f, 0.f, 0.f, 0.f, 0.f};
#pragma unroll
    for (int g = 0; g < 4; ++g) {
      float s = valid ? (S[g] * SCALE_) : -INFINITY;
      float rmax = s;                             // 16-lane half reduction
      rmax = fmaxf(rmax, __shfl_xor(rmax, 1, 32));
      rmax = fmaxf(rmax, __shfl_xor(rmax, 2, 32));
      rmax = fmaxf(rmax, __shfl_xor(rmax, 4, 32));
      rmax = fmaxf(rmax, __shfl_xor(rmax, 8, 32));
      const float mnew  = fmaxf(mrow[g], rmax);
      const float alpha = __expf(mrow[g] - mnew);
      const float p     = __expf(s - mnew);
      float rsum = p;
      rsum += __shfl_xor(rsum, 1, 32);
      rsum += __shfl_xor(rsum, 2, 32);
      rsum += __shfl_xor(rsum, 4, 32);
      rsum += __shfl_xor(rsum, 8, 32);
      lrow[g] = lrow[g] * alpha + rsum;
      mrow[g] = mnew;
      const float asc = (half == 0) ? alpha : 1.0f;
#pragma unroll
      for (int nt = 0; nt < 8; ++nt) acc[nt][g] *= asc;
      pvals[g] = p;
    }

    // ---- relayout P (C layout -> A layout) through per-wave LDS ----
#pragma unroll
    for (int g = 0; g < 8; ++g) {
      const int row = g + 8 * half;
      ldsP[w][row][m16] = (row < G_) ? pvals[g] : 0.0f;
    }
    asm volatile("s_wait_dscnt 0" ::: "memory");  // per-wave DS RAW fence
    v2f pa[4];
#pragma unroll
    for (int c = 0; c < 4; ++c) {
      pa[c].x = ldsP[w][m16][4 * c + koff];
      pa[c].y = ldsP[w][m16][4 * c + koff + 1];
    }

    // ---- O += P @ V, 8 N-tiles of 16 over D=128 ----
#pragma unroll
    for (int c = 0; c < 4; ++c) {
      const int ta = t0 + 4 * c + koff;
      const float* vra = kv_row(vcache, vnrow, bt, ta,     pos, h);
      const float* vrb = kv_row(vcache, vnrow, bt, ta + 1, pos, h);
#pragma unroll
      for (int nt = 0; nt < 8; ++nt) {
        v2f vb;
        vb.x = vra[nt * 16 + m16];
        vb.y = vrb[nt * 16 + m16];
        acc[nt] = wmma_f32_4(pa[c], vb, acc[nt]);
      }
    }
  }

  // ---- write per-wave partials to LDS ----
  if (half == 0) {
#pragma unroll
    for (int g = 0; g < 4; ++g)
#pragma unroll
      for (int nt = 0; nt < 8; ++nt)
        ldsO[w][g][nt * 16 + m16] = acc[nt][g];
    if (lane == 0) {
#pragma unroll
      for (int g = 0; g < 4; ++g) { ldsM[w][g] = mrow[g]; ldsL[w][g] = lrow[g]; }
    }
  }
  __syncthreads();

  // ---- combine 8 waves, normalize, write out ----
  for (int idx = tid; idx < G_ * D_; idx += 256) {
    const int r = idx >> 7;          // group row 0..3
    const int d = idx & 127;
    float M = -INFINITY;
#pragma unroll
    for (int ww = 0; ww < 8; ++ww) M = fmaxf(M, ldsM[ww][r]);
    float L = 0.f, O = 0.f;
#pragma unroll
    for (int ww = 0; ww < 8; ++ww) {
      const float f = __expf(ldsM[ww][r] - M);
      L += f * ldsL[ww][r];
      O += f * ldsO[ww][r][d];
    }
    out[b * (H_ * D_) + (h * G_ + r) * D_ + d] = O / L;
  }
}

// ---------------------------------------------------------------------------
extern "C" void kernel_launch(void* const* d_in, const int* in_sizes, int n_in,
                              void* d_out, int out_size, void* d_ws, size_t ws_size,
                              hipStream_t stream) {
  const float* x      = (const float*)d_in[0];   // [16,1,4096]
  const float* wq     = (const float*)d_in[1];   // [4096,4096]
  const float* wk     = (const float*)d_in[2];   // [4096,1024]
  const float* wv     = (const float*)d_in[3];   // [4096,1024]
  const float* kcache = (const float*)d_in[4];   // [1024,8,64,128]
  const float* vcache = (const float*)d_in[5];
  const int*   bt     = (const int*)d_in[6];     // [16,64]
  const int*   lens   = (const int*)d_in[7];     // [16]
  float* out = (float*)d_out;                    // [16,1,4096]

  float* qws  = (float*)d_ws;                    // 16*32*128
  float* knws = qws + B_ * H_ * D_;              // 16*8*128
  float* vnws = knws + B_ * KVH_ * D_;           // 16*8*128

  qkv_kernel<<<(H_ + 2 * KVH_) * D_ / 16, 32, 0, stream>>>(x, wq, wk, wv,
                                                           qws, knws, vnws);
  rope_kernel<<<B_ * (H_ + KVH_), 64, 0, stream>>>(qws, knws, lens);
  attn_kernel<<<B_ * KVH_, 256, 0, stream>>>(qws, knws, vnws,
                                             kcache, vcache, bt, lens, out);
}